// computeAllAtomCoordinates_7413113553276
// MI455X (gfx1250) — compile-verified
//
#include <hip/hip_runtime.h>
#include <stdint.h>
#include <math.h>

typedef __attribute__((ext_vector_type(2))) float v2f;
typedef __attribute__((ext_vector_type(8))) float v8f;

// ---------------------------------------------------------------------------
// Per-frame constant transforms.
// ---------------------------------------------------------------------------
constexpr float TRc[7][3][3] = {
    {{0.98f, 0.15f, -0.127f}, {-0.15f, 0.988f, 0.04f}, {0.127f, -0.04f, 0.991f}},
    {{0.992f, -0.11f, 0.045f}, {0.109f, 0.994f, 0.026f}, {-0.046f, -0.025f, 0.998f}},
    {{0.987f, 0.143f, -0.064f}, {-0.143f, 0.989f, 0.029f}, {0.065f, -0.028f, 0.997f}},
    {{0.994f, 0.108f, 0.012f}, {-0.108f, 0.994f, -0.032f}, {-0.012f, 0.032f, 1.0f}},
    {{0.999f, 0.021f, -0.034f}, {-0.021f, 0.998f, 0.055f}, {0.034f, -0.055f, 0.998f}},
    {{0.985f, -0.171f, 0.04f}, {0.17f, 0.985f, 0.043f}, {-0.041f, -0.042f, 0.999f}},
    {{0.992f, -0.125f, 0.011f}, {0.124f, 0.989f, -0.073f}, {-0.02f, 0.071f, 0.997f}},
};
constexpr float TTc[7][3] = {
    {0.78f, 0.34f, 0.26f}, {0.65f, 0.2f, -0.05f}, {0.72f, 0.25f, -0.08f},
    {1.3f, 0.25f, -0.1f},  {1.1f, 0.2f, -0.08f}, {1.24f, 0.3f, -0.05f},
    {1.05f, 0.15f, -0.12f},
};

// ---------------------------------------------------------------------------
// Constant A-operands for V_WMMA_F32_16X16X4_F32 (16 feature rows x K=4).
// Slot scheme per 8-row half h (owned by lane half), sub-slot s in {0,1}:
//   rows h*8+s*3+j (j=0..2) : A'[.,j] features of frame fr[h*2+s]  (TR[f][k][j])
//   row  h*8+6+s            : R@TRANS_T feature of frame fr[h*2+s] (TT[f][k])
// K=3 column is zero-padded.
// ---------------------------------------------------------------------------
struct AMat { float v[16][4]; };

constexpr AMat buildA(int fa, int fb, int fc, int fd) {
    AMat a{};
    const int fr[4] = {fa, fb, fc, fd};
    for (int h = 0; h < 2; ++h) {
        for (int s = 0; s < 2; ++s) {
            const int f = fr[h * 2 + s];
            if (f < 0) continue;
            for (int j = 0; j < 3; ++j)
                for (int k = 0; k < 3; ++k)
                    a.v[h * 8 + s * 3 + j][k] = TRc[f][k][j];
            for (int k = 0; k < 3; ++k)
                a.v[h * 8 + 6 + s][k] = TTc[f][k];
        }
    }
    return a;
}

constexpr AMat A1m = buildA(0, 1, 2, 3);   // WMMA #1: frames 0,1 (half0), 2,3 (half1)
constexpr AMat A2m = buildA(4, 5, 6, -1);  // WMMA #2: frames 4,5 (half0), 6 (half1)
constexpr int frameTab[2][2][2] = {{{0, 1}, {2, 3}}, {{4, 5}, {6, -1}}};

// ---------------------------------------------------------------------------
// CDNA5 async global<->LDS helpers (ASYNCcnt-tracked, 16B per enabled lane).
// ---------------------------------------------------------------------------
__device__ __forceinline__ void async_load_b128(uint32_t lds_byte_addr, const void* gaddr) {
    asm volatile("global_load_async_to_lds_b128 %0, %1, off"
                 :: "v"(lds_byte_addr), "v"(gaddr)
                 : "memory");
}
__device__ __forceinline__ void async_store_b128(void* gaddr, uint32_t lds_byte_addr) {
    asm volatile("global_store_async_from_lds_b128 %0, %1, off"
                 :: "v"(gaddr), "v"(lds_byte_addr)
                 : "memory");
}
__device__ __forceinline__ void wait_async0() {
    asm volatile("s_wait_asynccnt 0" ::: "memory");
}

// ---------------------------------------------------------------------------
// Scalar per-residue math (tail path only).
// ---------------------------------------------------------------------------
__device__ __forceinline__ void compute_res(const float Rm[9], const float tv[3],
                                            const float* __restrict__ av,
                                            float* __restrict__ orot,
                                            float* __restrict__ otr) {
#pragma unroll
    for (int f = 0; f < 7; ++f) {
        const float a0 = av[f * 4 + 0], a1 = av[f * 4 + 1];
        const float a2 = av[f * 4 + 2], a3 = av[f * 4 + 3];
        const float inv = 1.0f / sqrtf(a0 * a0 + a1 * a1 + a2 * a2 + a3 * a3);
        const float c = a2 * inv, s = a3 * inv;
        float A[3][3];
#pragma unroll
        for (int i = 0; i < 3; ++i)
#pragma unroll
            for (int j = 0; j < 3; ++j)
                A[i][j] = Rm[i * 3 + 0] * TRc[f][0][j] + Rm[i * 3 + 1] * TRc[f][1][j]
                        + Rm[i * 3 + 2] * TRc[f][2][j];
#pragma unroll
        for (int i = 0; i < 3; ++i) {
            orot[f * 9 + i * 3 + 0] = A[i][0];
            orot[f * 9 + i * 3 + 1] = c * A[i][1] + s * A[i][2];
            orot[f * 9 + i * 3 + 2] = c * A[i][2] - s * A[i][1];
            otr[f * 3 + i] = Rm[i * 3 + 0] * TTc[f][0] + Rm[i * 3 + 1] * TTc[f][1]
                           + Rm[i * 3 + 2] * TTc[f][2] + tv[i];
        }
    }
}

// ---------------------------------------------------------------------------
// Kernel: 128 residues/block (4 waves x 32 residues). Inputs async-staged to
// LDS; per wave, residue rows are packed into V_WMMA_F32_16X16X4_F32 B
// operands against constant A operands; lane-local RotX recombination; LDS
// output image async-streamed to global with coalesced B128 stores.
//
// LDS float layout (15872 floats = 62 KB, 16B-aligned regions):
//   [    0,  1152)  R tile     [ 1152, 1536)  t tile   [ 1536, 5120) angles
//   [ 5120, 13184)  R_out tile [13184,15872)  t_out tile
// ---------------------------------------------------------------------------
#define TILE 128

__global__ __launch_bounds__(TILE) void frames_kernel(
    const float* __restrict__ R, const float* __restrict__ t,
    const float* __restrict__ ang, float* __restrict__ out, int Ntot) {
    __shared__ __align__(16) float smem[15872];

    const int tid = threadIdx.x;
    const int n0 = blockIdx.x * TILE;

    float* sR = smem;            // 1152 floats (4608 B  = 288 B128)
    float* sT = smem + 1152;     //  384 floats (1536 B  =  96 B128)
    float* sA = smem + 1536;     // 3584 floats (14336 B = 896 B128)
    float* sOR = smem + 5120;    // 8064 floats (32256 B = 2016 B128)
    float* sOT = smem + 13184;   // 2688 floats (10752 B = 672 B128)

    const bool fast = (n0 + TILE <= Ntot) && ((Ntot & 3) == 0);

    if (fast) {
        const uint32_t lbase = (uint32_t)(uintptr_t)(void*)smem;
        const char* gR = (const char*)(R + (size_t)n0 * 9);
        const char* gT = (const char*)(t + (size_t)n0 * 3);
        const char* gA = (const char*)(ang + (size_t)n0 * 28);

        // ---- async stage inputs into LDS (coalesced B128) ----
        for (int i = tid; i < 288; i += TILE)
            async_load_b128(lbase + 0u + (uint32_t)i * 16u, gR + (size_t)i * 16);
        if (tid < 96)
            async_load_b128(lbase + 4608u + (uint32_t)tid * 16u, gT + (size_t)tid * 16);
#pragma unroll
        for (int k = 0; k < 7; ++k) {
            const int i = tid + k * TILE;
            async_load_b128(lbase + 6144u + (uint32_t)i * 16u, gA + (size_t)i * 16);
        }
        wait_async0();
        __syncthreads();

        // ---- WMMA compute: wave handles 32 residues = 96 stacked R-rows ----
        const int lane = tid & 31;
        const int wb = (tid >> 5) * 32;   // wave residue base within tile
        const int m = lane & 15;          // feature row / column index
        const int hi = lane >> 4;         // which 8-row D half / K half

        // Constant A operands (ISA 32-bit 16x4 layout: VGPR0 = K0|K2, VGPR1 = K1|K3)
        v2f a1, a2;
        a1[0] = A1m.v[m][hi ? 2 : 0]; a1[1] = A1m.v[m][hi ? 3 : 1];
        a2[0] = A2m.v[m][hi ? 2 : 0]; a2[1] = A2m.v[m][hi ? 3 : 1];
        const v8f cz = {};

#pragma unroll
        for (int b = 0; b < 6; ++b) {
            const int r = b * 16 + m;     // local stacked row 0..95
            const int nl = r / 3;         // local residue 0..31
            const int i = r - nl * 3;     // row of R within residue
            const int rbase = (wb + nl) * 9 + i * 3;

            // B operand (4x16): lanes 0-15 supply K0,K1; lanes 16-31 supply K2,pad0
            v2f bop;
            bop[0] = sR[rbase + (hi ? 2 : 0)];
            bop[1] = hi ? 0.0f : sR[rbase + 1];

            v8f d1 = __builtin_amdgcn_wmma_f32_16x16x4_f32(
                false, a1, false, bop, (short)0, cz, false, false);
            v8f d2 = __builtin_amdgcn_wmma_f32_16x16x4_f32(
                false, a2, false, bop, (short)0, cz, false, false);

            const float tvi = sT[(wb + nl) * 3 + i];
            const int abase = (wb + nl) * 28;
            const int orbase = (wb + nl) * 63 + i * 3;
            const int otbase = (wb + nl) * 21 + i;

#pragma unroll
            for (int w = 0; w < 2; ++w) {
                const v8f d = w ? d2 : d1;
#pragma unroll
                for (int s = 0; s < 2; ++s) {
                    const int f = frameTab[w][hi][s];
                    if (f >= 0) {
                        const float p0 = d[s * 3 + 0];
                        const float p1 = d[s * 3 + 1];
                        const float p2 = d[s * 3 + 2];
                        const float pt = d[6 + s];
                        const float x0 = sA[abase + f * 4 + 0];
                        const float x1 = sA[abase + f * 4 + 1];
                        const float x2 = sA[abase + f * 4 + 2];
                        const float x3 = sA[abase + f * 4 + 3];
                        const float inv =
                            1.0f / sqrtf(x0 * x0 + x1 * x1 + x2 * x2 + x3 * x3);
                        const float cc = x2 * inv, ss = x3 * inv;
                        sOR[orbase + f * 9 + 0] = p0;
                        sOR[orbase + f * 9 + 1] = cc * p1 + ss * p2;
                        sOR[orbase + f * 9 + 2] = cc * p2 - ss * p1;
                        sOT[otbase + f * 3] = pt + tvi;
                    }
                }
            }
        }
        __syncthreads();

        // ---- async stream outputs LDS -> global (coalesced B128) ----
        char* goR = (char*)(out + (size_t)n0 * 63);
        char* goT = (char*)(out + (size_t)Ntot * 63 + (size_t)n0 * 21);
        for (int i = tid; i < 2016; i += TILE)
            async_store_b128(goR + (size_t)i * 16, lbase + 20480u + (uint32_t)i * 16u);
        for (int i = tid; i < 672; i += TILE)
            async_store_b128(goT + (size_t)i * 16, lbase + 52736u + (uint32_t)i * 16u);
        wait_async0();
    } else {
        // Tail / alignment fallback: plain per-thread path (no WMMA: partial EXEC).
        const int n = n0 + tid;
        if (n < Ntot) {
            float Rm[9], tv[3], av[28], orot[63], otr[21];
#pragma unroll
            for (int i = 0; i < 9; ++i) Rm[i] = R[(size_t)n * 9 + i];
#pragma unroll
            for (int i = 0; i < 3; ++i) tv[i] = t[(size_t)n * 3 + i];
#pragma unroll
            for (int i = 0; i < 28; ++i) av[i] = ang[(size_t)n * 28 + i];
            compute_res(Rm, tv, av, orot, otr);
            float* oR = out + (size_t)n * 63;
            float* oT = out + (size_t)Ntot * 63 + (size_t)n * 21;
#pragma unroll
            for (int i = 0; i < 63; ++i) oR[i] = orot[i];
#pragma unroll
            for (int i = 0; i < 21; ++i) oT[i] = otr[i];
        }
    }
}

// ---------------------------------------------------------------------------
// Launch. Inputs: d_in[0]=R [N,3,3], d_in[1]=t [N,3], d_in[2]=angles [N,7,2,2];
// out = [R_out (N*63) | t_out (N*21)] floats.
// ---------------------------------------------------------------------------
extern "C" void kernel_launch(void* const* d_in, const int* in_sizes, int n_in,
                              void* d_out, int out_size, void* d_ws, size_t ws_size,
                              hipStream_t stream) {
    (void)n_in; (void)out_size; (void)d_ws; (void)ws_size;
    const float* R = (const float*)d_in[0];
    const float* t = (const float*)d_in[1];
    const float* ang = (const float*)d_in[2];
    float* out = (float*)d_out;
    const int Ntot = in_sizes[0] / 9;
    const int nblocks = (Ntot + TILE - 1) / TILE;
    hipLaunchKernelGGL(frames_kernel, dim3(nblocks), dim3(TILE), 0, stream,
                       R, t, ang, out, Ntot);
}